// sLSTM_14903536517286
// MI455X (gfx1250) — compile-verified
//
#include <hip/hip_runtime.h>
#include <hip/hip_bf16.h>
#include <math.h>

// sLSTM cell, fused: 8 GEMMs (K = 2*4096, A=[x|h^T], B_g=[W_g;U_g]) + epilogue.
// fp32 WMMA (V_WMMA_F32_16X16X4_F32), double-buffered LDS via
// global_load_async_to_lds_b128 (ASYNCcnt), wave32, 256-thread WGs.

#define DDIM 4096
#define BDIM 4096
#define KT   32            // K tile
#define WGM  64            // M (batch) per workgroup
#define WGN  128           // N (hidden) per workgroup

// LDS layout (floats), per buffer:
//  A region: x-phase  Axk[64][36] (m-major, padded)  OR  h-phase Akm[32][64] (k-major)
//  B region: 4 gates * [32][128]
#define AXK_STRIDE 36
#define A_FLOATS   2304                       // max(64*36, 32*64)
#define BOFF       A_FLOATS
#define BGATE      (KT * WGN)                 // 4096 floats per gate
#define BUF_FLOATS (A_FLOATS + 4 * BGATE)     // 18688 floats = 74752 B
#define NBUF       2
#define ASYNC_PER_THREAD 18                   // 2 (A) + 16 (B) per tile

typedef float v2f __attribute__((ext_vector_type(2)));
typedef float v8f __attribute__((ext_vector_type(8)));

__device__ __forceinline__ void async_copy16(const float* gsrc, unsigned lds_byte_off) {
  // copies 16 bytes from the global address gsrc to the LDS byte address lds_byte_off, tracked by ASYNCcnt.
  asm volatile("global_load_async_to_lds_b128 %0, %1, off"
               :: "v"(lds_byte_off),
                  "v"((unsigned long long)(uintptr_t)gsrc)
               : "memory");
}

extern "C" __global__ __launch_bounds__(256)
void slstm_fused_kernel(const float* __restrict__ x,  const float* __restrict__ h,
                        const float* __restrict__ c,  const float* __restrict__ m,
                        const float* __restrict__ nrm,
                        const float* __restrict__ wi, const float* __restrict__ wf,
                        const float* __restrict__ wo, const float* __restrict__ wc,
                        const float* __restrict__ ui, const float* __restrict__ uf,
                        const float* __restrict__ uo, const float* __restrict__ uc,
                        const float* __restrict__ bi, const float* __restrict__ bf,
                        const float* __restrict__ bo, const float* __restrict__ bc,
                        float* __restrict__ out)
{
  extern __shared__ float smem[];
  const int tid  = threadIdx.x;
  const int lane = tid & 31;
  const int wave = tid >> 5;       // 8 waves
  const int wm   = wave & 1;       // 2 wave rows  (32 M each)
  const int wn   = wave >> 1;      // 4 wave cols  (32 N each)
  const int m0   = blockIdx.x * WGM;   // M-blocks fastest -> weight panels shared via L2
  const int n0   = blockIdx.y * WGN;

  const float* Wg[4]   = {wi, wf, wo, wc};
  const float* Ug[4]   = {ui, uf, uo, uc};
  const float* Bias[4] = {bi, bf, bo, bc};

  v8f acc[4][2][2];
  {
    v8f z = {};
    #pragma unroll
    for (int g = 0; g < 4; ++g)
      #pragma unroll
      for (int i = 0; i < 2; ++i)
        #pragma unroll
        for (int j = 0; j < 2; ++j)
          acc[g][i][j] = z;
  }

  // ---- async producer: stage one K-tile into LDS buffer `buf` ----
  auto issue = [&](int kt, int buf) {
    const int k0 = kt * KT;
    const bool xreg = (k0 < DDIM);
    const unsigned base = (unsigned)(buf * BUF_FLOATS) * 4u;

    if (xreg) {
      // A <- x rows: Axk[row][k], padded stride 36 (144B, 16B aligned)
      #pragma unroll
      for (int t = 0; t < 2; ++t) {
        int cid = tid * 2 + t;                  // 0..511
        int row = cid >> 3;
        int c4  = (cid & 7) * 4;
        const float* g = x + (size_t)(m0 + row) * DDIM + (k0 + c4);
        async_copy16(g, base + (unsigned)(row * AXK_STRIDE + c4) * 4u);
      }
    } else {
      // A <- h rows (contiguous in batch): Akm[k][b]
      #pragma unroll
      for (int t = 0; t < 2; ++t) {
        int cid = tid * 2 + t;                  // 0..511
        int kr  = cid >> 4;
        int c4  = (cid & 15) * 4;
        const float* g = h + (size_t)(k0 - DDIM + kr) * BDIM + (m0 + c4);
        async_copy16(g, base + (unsigned)(kr * WGM + c4) * 4u);
      }
    }

    const int krb = xreg ? k0 : (k0 - DDIM);
    #pragma unroll
    for (int t = 0; t < 16; ++t) {
      int g4 = t >> 2;                          // gate
      int kr = (tid >> 5) + (t & 3) * 8;        // 0..31
      int c4 = (tid & 31) * 4;                  // 0..124
      const float* wp = (xreg ? Wg[g4] : Ug[g4]) + (size_t)(krb + kr) * DDIM + (n0 + c4);
      async_copy16(wp, base + (unsigned)(BOFF + g4 * BGATE + kr * WGN + c4) * 4u);
    }
  };

  // ---- consumer: 8 WMMA k-steps over one K-tile ----
  auto compute = [&](int kt, int buf) {
    const bool xreg = (kt * KT < DDIM);
    const float* sb = smem + buf * BUF_FLOATS;
    const int koff = (lane >> 4) * 2;   // lanes 16-31 hold K+2/K+3 halves
    const int ml   = lane & 15;

    #pragma unroll
    for (int kk = 0; kk < KT; kk += 4) {
      v2f a[2];
      if (xreg) {
        #pragma unroll
        for (int ti = 0; ti < 2; ++ti) {
          const float* p = sb + (wm * 32 + ti * 16 + ml) * AXK_STRIDE + kk + koff;
          a[ti].x = p[0];
          a[ti].y = p[1];
        }
      } else {
        #pragma unroll
        for (int ti = 0; ti < 2; ++ti) {
          int col = wm * 32 + ti * 16 + ml;
          a[ti].x = sb[(kk + koff    ) * WGM + col];
          a[ti].y = sb[(kk + koff + 1) * WGM + col];
        }
      }
      #pragma unroll
      for (int g = 0; g < 4; ++g) {
        #pragma unroll
        for (int tj = 0; tj < 2; ++tj) {
          int ncol = wn * 32 + tj * 16 + ml;
          v2f b;
          b.x = sb[BOFF + g * BGATE + (kk + koff    ) * WGN + ncol];
          b.y = sb[BOFF + g * BGATE + (kk + koff + 1) * WGN + ncol];
          #pragma unroll
          for (int ti = 0; ti < 2; ++ti) {
            acc[g][ti][tj] = __builtin_amdgcn_wmma_f32_16x16x4_f32(
                false, a[ti], false, b, (short)0, acc[g][ti][tj], false, false);
          }
        }
      }
    }
  };

  // ---- main K loop, double buffered ----
  const int NKT = (2 * DDIM) / KT;  // 256
  issue(0, 0);
  for (int kt = 0; kt < NKT; ++kt) {
    const int buf = kt & 1;
    if (kt + 1 < NKT) {
      issue(kt + 1, buf ^ 1);
      asm volatile("s_wait_asynccnt 18" ::: "memory");   // tile kt fully landed
    } else {
      asm volatile("s_wait_asynccnt 0" ::: "memory");
    }
    __syncthreads();
    compute(kt, buf);
    __syncthreads();            // readers done before next overwrite of buf
  }

  // ---- epilogue: sLSTM recurrence ----
  const size_t BD = (size_t)BDIM * DDIM;
  const int ml2 = lane & 15;
  const int rh  = (lane >> 4) * 8;

  float bias[4][2];
  #pragma unroll
  for (int g = 0; g < 4; ++g)
    #pragma unroll
    for (int tj = 0; tj < 2; ++tj)
      bias[g][tj] = Bias[g][n0 + wn * 32 + tj * 16 + ml2];

  #pragma unroll
  for (int ti = 0; ti < 2; ++ti) {
    #pragma unroll
    for (int tj = 0; tj < 2; ++tj) {
      #pragma unroll
      for (int r = 0; r < 8; ++r) {
        const int row = m0 + wm * 32 + ti * 16 + rh + r;
        const int col = n0 + wn * 32 + tj * 16 + ml2;
        const size_t idx = (size_t)row * DDIM + col;

        const float ai = acc[0][ti][tj][r] + bias[0][tj];
        const float af = acc[1][ti][tj][r] + bias[1][tj];
        const float ao = acc[2][ti][tj][r] + bias[2][tj];
        const float az = acc[3][ti][tj][r] + bias[3][tj];

        const float mv = m[idx], cv = c[idx], nv = nrm[idx];
        const float m_new = fmaxf(af + mv, ai);
        const float i_p   = expf(ai - m_new);
        const float f_p   = expf(af);                 // faithful: +m_new-m_new cancels
        const float o_t   = 1.f / (1.f + expf(-ao));
        const float z_t   = tanhf(az);
        const float c_new = f_p * cv + i_p * z_t;
        const float n_new = f_p * nv + i_p;
        const float h_new = o_t * (c_new / n_new);

        out[idx]          = h_new;
        out[BD + idx]     = c_new;
        out[2 * BD + idx] = m_new;
        out[3 * BD + idx] = n_new;
      }
    }
  }
}

extern "C" void kernel_launch(void* const* d_in, const int* in_sizes, int n_in,
                              void* d_out, int out_size, void* d_ws, size_t ws_size,
                              hipStream_t stream) {
  (void)in_sizes; (void)n_in; (void)out_size; (void)d_ws; (void)ws_size;
  const float* x  = (const float*)d_in[0];
  const float* h  = (const float*)d_in[1];
  const float* c  = (const float*)d_in[2];
  const float* m  = (const float*)d_in[3];
  const float* n  = (const float*)d_in[4];
  const float* wi = (const float*)d_in[5];
  const float* wf = (const float*)d_in[6];
  const float* wo = (const float*)d_in[7];
  const float* wc = (const float*)d_in[8];
  const float* ui = (const float*)d_in[9];
  const float* uf = (const float*)d_in[10];
  const float* uo = (const float*)d_in[11];
  const float* uc = (const float*)d_in[12];
  const float* bi = (const float*)d_in[13];
  const float* bf = (const float*)d_in[14];
  const float* bo = (const float*)d_in[15];
  const float* bc = (const float*)d_in[16];
  float* out = (float*)d_out;

  dim3 grid(BDIM / WGM, DDIM / WGN);   // (64, 32): M fastest for L2 weight reuse
  dim3 block(256);
  size_t shmem = (size_t)NBUF * BUF_FLOATS * sizeof(float);  // 149504 B
  hipLaunchKernelGGL(slstm_fused_kernel, grid, block, shmem, stream,
                     x, h, c, m, n, wi, wf, wo, wc, ui, uf, uo, uc,
                     bi, bf, bo, bc, out);
}